// Paillier_Pooling_84370337562660
// MI455X (gfx1250) — compile-verified
//
#include <hip/hip_runtime.h>

// Paillier 2x2 sum-pool == windowed modular product mod NSQ = 46337^2.
// Memory-bound: 1.25 GiB traffic -> ~58 us floor at 23.3 TB/s.
// Strategy: TDM (tensor_load_to_lds) stages 2-row input tiles into LDS,
// compute 3 modmuls per output, nontemporal b128 stores.

typedef unsigned long long u64;
typedef unsigned int       u32;
typedef u32 v4u  __attribute__((ext_vector_type(4)));
typedef int v4i  __attribute__((ext_vector_type(4)));
typedef int v8i  __attribute__((ext_vector_type(8)));
typedef u64 v2u64 __attribute__((ext_vector_type(2)));

#define NSQ_C 2147117569ULL   // 46337^2, fits in 31 bits; a*b < 2^62

#define B_   32
#define H_   256
#define W_   256
#define C_   64
#define HO_  128
#define WO_  128
#define ROW_ELEMS   (W_ * C_)            // 16384 int64 per input row
#define TILE_WCOLS  32                    // input w-columns per block
#define TILE_ELEMS  (TILE_WCOLS * C_)     // 2048 int64 per row-slice
#define WO_PER_BLOCK (TILE_WCOLS / 2)     // 16 output columns per block
#define CHUNKS      (WO_ / WO_PER_BLOCK)  // 8 w-chunks

__device__ __forceinline__ u64 mulmod(u64 a, u64 b) {
    // a,b < NSQ < 2^31 so a*b < 2^62: no overflow; constant divisor -> magic mul.
    return (a * b) % NSQ_C;
}

__global__ __launch_bounds__(256)
void pail_pool_kernel(const u64* __restrict__ x, u64* __restrict__ out) {
    __shared__ u64 smem[2 * TILE_ELEMS];   // 32 KB: rows h and h+1

    const u32 chunk = blockIdx.x & (CHUNKS - 1);
    const u32 bh    = blockIdx.x >> 3;
    const u32 ho    = bh & (HO_ - 1);
    const u32 b     = bh >> 7;

    // element offset of tile start: (b, h=2*ho, w=chunk*32, c=0)
    const u64 tileStart = ((u64)(b * H_ + 2u * ho) * (u64)W_ + (u64)(chunk * TILE_WCOLS)) * (u64)C_;

#if __has_builtin(__builtin_amdgcn_tensor_load_to_lds) && __has_builtin(__builtin_amdgcn_s_wait_tensorcnt)
    // --- CDNA5 Tensor Data Mover path: one 2D descriptor per workgroup ---
    if (threadIdx.x < 32u) {               // first wave issues the DMA
        const u64 gaddr    = (u64)(const void*)(x + tileStart);         // byte address
        const u32 lds_base = (u32)(u64)(const void*)&smem[0];           // LDS byte offset

        v4u g0; v8i g1; v4i g2; v4i g3; v8i g4;
        // D# group 0 (ISA 8.3): [1:0]=count=1, [63:32]=lds_addr,
        // [120:64]=global_addr, [127:126]=type=2
        g0[0] = 1u;
        g0[1] = lds_base;
        g0[2] = (u32)gaddr;
        g0[3] = ((u32)(gaddr >> 32) & 0x01FFFFFFu) | (2u << 30);
        // D# group 1 (ISA 8.4): data_size=3 (8B); tensor_dim0=ROW_ELEMS;
        // tensor_dim1=2; tile_dim0=TILE_ELEMS; tile_dim1=2; tile_dim2=0;
        // tensor_dim0_stride=ROW_ELEMS; no padding/iterate/multicast.
        g1[0] = (int)(3u << 16);                               // data_size = 8 bytes
        g1[1] = (int)((u32)(ROW_ELEMS & 0xFFFF) << 16);        // tensor_dim0[15:0]
        g1[2] = (int)(((u32)ROW_ELEMS >> 16) | (2u << 16));    // tensor_dim0[31:16] | tensor_dim1
        g1[3] = (int)((u32)TILE_ELEMS << 16);                  // tile_dim0
        g1[4] = 2;                                             // tile_dim1 = 2 rows
        g1[5] = ROW_ELEMS;                                     // tensor_dim0_stride[31:0]
        g1[6] = 0;
        g1[7] = 0;
        g2[0] = g2[1] = g2[2] = g2[3] = 0;                     // 2D tile: groups 2/3 unused
        g3[0] = g3[1] = g3[2] = g3[3] = 0;
        g4[0] = g4[1] = g4[2] = g4[3] = 0;                     // extra group (6-arg builtin), zero
        g4[4] = g4[5] = g4[6] = g4[7] = 0;

        // 6-arg variant (clang-23 / therock-10.0 headers):
        // (uint32x4, int32x8, int32x4, int32x4, int32x8, i32 cpol)
        __builtin_amdgcn_tensor_load_to_lds(g0, g1, g2, g3, g4, 0);
        __builtin_amdgcn_s_wait_tensorcnt(0);
    }
#else
    // Fallback: cooperative vector loads into LDS
    for (u32 i = threadIdx.x; i < (u32)TILE_ELEMS; i += 256u) {
        smem[i]              = x[tileStart + i];
        smem[TILE_ELEMS + i] = x[tileStart + ROW_ELEMS + i];
    }
#endif
    __syncthreads();

    // 16 wo x 64 c outputs per block = 512 v2u64 stores; 2 per thread.
    const u32 cp  = threadIdx.x & 31u;     // channel pair
    const u32 c   = cp << 1;
    const u32 woB = threadIdx.x >> 5;      // 0..7
    const u64 outRow = (u64)(b * HO_ + ho) * (u64)WO_ + (u64)(chunk * WO_PER_BLOCK);

#pragma unroll
    for (int i = 0; i < 2; ++i) {
        const u32 wo = woB + (u32)(i << 3);
        const u64* p0 = &smem[(2u * wo) * C_ + c];       // row h,  col 2*wo
        const u64* p1 = p0 + TILE_ELEMS;                 // row h+1
        const v2u64 a0 = *(const v2u64*)(p0);            // ds_load_b128
        const v2u64 a1 = *(const v2u64*)(p0 + C_);       // col 2*wo+1
        const v2u64 b0 = *(const v2u64*)(p1);
        const v2u64 b1 = *(const v2u64*)(p1 + C_);
        v2u64 r;
        r.x = mulmod(mulmod(mulmod(a0.x, a1.x), b0.x), b1.x);
        r.y = mulmod(mulmod(mulmod(a0.y, a1.y), b0.y), b1.y);
        __builtin_nontemporal_store(r, (v2u64*)(out + (outRow + wo) * (u64)C_ + c));
    }
}

extern "C" void kernel_launch(void* const* d_in, const int* in_sizes, int n_in,
                              void* d_out, int out_size, void* d_ws, size_t ws_size,
                              hipStream_t stream) {
    (void)in_sizes; (void)n_in; (void)out_size; (void)d_ws; (void)ws_size;
    const u64* x  = (const u64*)d_in[0];   // int64 ciphertexts, values in [0, NSQ)
    u64*      out = (u64*)d_out;           // int64 output
    dim3 grid(B_ * HO_ * CHUNKS);          // 32 * 128 * 8 = 32768 blocks
    pail_pool_kernel<<<grid, 256, 0, stream>>>(x, out);
}